// HomoCONV_90091234001075
// MI455X (gfx1250) — compile-verified
//
#include <hip/hip_runtime.h>
#include <hip/hip_bf16.h>

typedef __attribute__((ext_vector_type(2))) float v2f;
typedef __attribute__((ext_vector_type(8))) float v8f;

#define N_NODES 50000
#define N_EDGES 800000
#define HIDDEN  128
#define OUTF    10
#define N_GRAPHS 128
#define LN_EPS 1e-5f
#define MTILE 32   // rows per block (2 row tiles of 16 per wave)

// ---------------- degree ----------------
__global__ void deg_kernel(const int* __restrict__ dst, float* __restrict__ deg) {
    int e = blockIdx.x * blockDim.x + threadIdx.x;
    if (e < N_EDGES) atomicAdd(&deg[dst[e]], 1.0f);
}

__global__ void invdeg_kernel(float* __restrict__ deg) {
    int i = blockIdx.x * blockDim.x + threadIdx.x;
    if (i < N_NODES) {
        float d = deg[i];
        deg[i] = (d > 0.0f) ? (1.0f / d) : 0.0f;
    }
}

// ---------------- edge scatter: agg[dst] += x[src] ----------------
// one wave per edge, lane owns 4 features (float4 gather + 4 f32 atomics, resolve in L2)
__global__ __launch_bounds__(256) void scatter_kernel(const float* __restrict__ x,
                                                      const int* __restrict__ src,
                                                      const int* __restrict__ dst,
                                                      float* __restrict__ agg) {
    int gid  = blockIdx.x * blockDim.x + threadIdx.x;
    int e    = gid >> 5;
    int lane = gid & 31;
    if (e >= N_EDGES) return;
    int s = src[e];
    int d = dst[e];
    const float4 v = *(const float4*)(x + (size_t)s * HIDDEN + lane * 4);
    float* out = agg + (size_t)d * HIDDEN + lane * 4;
    atomicAdd(out + 0, v.x);
    atomicAdd(out + 1, v.y);
    atomicAdd(out + 2, v.z);
    atomicAdd(out + 3, v.w);
}

// ---------------- fused GraphConv: (agg*invdeg)@Wrel + brel + x@Wroot, LN, ReLU ----------
// 256 threads = 8 waves; block owns 32 rows (two 16-row tiles); wave w owns output
// columns [16w,16w+16). Each K-step loads B fragments ONCE and feeds both row tiles:
// 4 chained V_WMMA_F32_16X16X4_F32 per K-step per wave -> halves W_rel/W_root L2 traffic.
__global__ __launch_bounds__(256) void gconv_kernel(const float* __restrict__ xin,
                                                    const float* __restrict__ agg,
                                                    const float* __restrict__ invdeg,
                                                    const float* __restrict__ Wrel,
                                                    const float* __restrict__ brel,
                                                    const float* __restrict__ Wroot,
                                                    const float* __restrict__ gamma,
                                                    const float* __restrict__ beta,
                                                    float* __restrict__ xout) {
    __shared__ float shx[MTILE * HIDDEN];   // x tile (row-major)
    __shared__ float sha[MTILE * HIDDEN];   // agg*invdeg tile
    __shared__ float ssum[MTILE];
    __shared__ float ssq[MTILE];

    const int tid      = threadIdx.x;
    const int blockRow = blockIdx.x * MTILE;

    // cooperative coalesced LDS fill (4096 elems per matrix, 16 per thread).
    // Tail block clamps the row so every lane stays active (WMMA needs EXEC all-ones).
    for (int i = tid; i < MTILE * HIDDEN; i += 256) {
        int r  = i >> 7;
        int gr = blockRow + r;
        int rc = (gr < N_NODES) ? gr : 0;
        size_t g = (size_t)rc * HIDDEN + (i & 127);
        shx[i] = xin[g];
        sha[i] = agg[g] * invdeg[rc];
    }
    if (tid < MTILE) { ssum[tid] = 0.0f; ssq[tid] = 0.0f; }
    __syncthreads();

    const int wave = tid >> 5;
    const int lane = tid & 31;
    const int half = lane >> 4;     // 0: K+0/K+1,  1: K+2/K+3 (ISA A/B fragment split)
    const int lr   = lane & 15;
    const int col  = wave * 16 + lr;   // output column owned by this lane

    v8f c0 = {};   // rows [0,16) of the block
    v8f c1 = {};   // rows [16,32) of the block
    #pragma unroll 4
    for (int k = 0; k < HIDDEN; k += 4) {
        const int ka = k + 2 * half;
        // B fragments (shared by both row tiles)
        v2f bR, bT;
        bR.x = Wrel[(size_t)ka * HIDDEN + col];
        bR.y = Wrel[(size_t)(ka + 1) * HIDDEN + col];
        bT.x = Wroot[(size_t)ka * HIDDEN + col];
        bT.y = Wroot[(size_t)(ka + 1) * HIDDEN + col];
        if (k + 32 < HIDDEN) {   // L1 prefetch of W rows ahead (global_prefetch_b8)
            __builtin_prefetch(&Wrel[(size_t)(ka + 32) * HIDDEN + col], 0, 1);
            __builtin_prefetch(&Wroot[(size_t)(ka + 32) * HIDDEN + col], 0, 1);
        }
        // A fragments, row tile 0
        v2f aA0, aX0, aA1, aX1;
        aA0.x = sha[lr * HIDDEN + ka];
        aA0.y = sha[lr * HIDDEN + ka + 1];
        aX0.x = shx[lr * HIDDEN + ka];
        aX0.y = shx[lr * HIDDEN + ka + 1];
        // A fragments, row tile 1
        aA1.x = sha[(lr + 16) * HIDDEN + ka];
        aA1.y = sha[(lr + 16) * HIDDEN + ka + 1];
        aX1.x = shx[(lr + 16) * HIDDEN + ka];
        aX1.y = shx[(lr + 16) * HIDDEN + ka + 1];

        c0 = __builtin_amdgcn_wmma_f32_16x16x4_f32(false, aA0, false, bR, (short)0, c0, false, false);
        c0 = __builtin_amdgcn_wmma_f32_16x16x4_f32(false, aX0, false, bT, (short)0, c0, false, false);
        c1 = __builtin_amdgcn_wmma_f32_16x16x4_f32(false, aA1, false, bR, (short)0, c1, false, false);
        c1 = __builtin_amdgcn_wmma_f32_16x16x4_f32(false, aX1, false, bT, (short)0, c1, false, false);
    }

    // bias (same column for all accumulator rows)
    const float bias = brel[col];
    #pragma unroll
    for (int r = 0; r < 8; ++r) { c0[r] += bias; c1[r] += bias; }

    // cross-wave row statistics via LDS f32 atomics (ds_add_f32)
    #pragma unroll
    for (int r = 0; r < 8; ++r) {
        int row = r + 8 * half;          // C layout: VGPR r -> row r (+8 for upper half)
        atomicAdd(&ssum[row], c0[r]);
        atomicAdd(&ssq[row], c0[r] * c0[r]);
        atomicAdd(&ssum[row + 16], c1[r]);
        atomicAdd(&ssq[row + 16], c1[r] * c1[r]);
    }
    __syncthreads();

    const float g  = gamma[col];
    const float bb = beta[col];
    #pragma unroll
    for (int r = 0; r < 8; ++r) {
        int row0 = r + 8 * half;
        int row1 = row0 + 16;
        {
            float mu  = ssum[row0] * (1.0f / (float)HIDDEN);
            float var = ssq[row0] * (1.0f / (float)HIDDEN) - mu * mu;
            float v   = g * (c0[r] - mu) * rsqrtf(var + LN_EPS) + bb;
            int grow  = blockRow + row0;
            if (grow < N_NODES)
                xout[(size_t)grow * HIDDEN + col] = fmaxf(v, 0.0f);
        }
        {
            float mu  = ssum[row1] * (1.0f / (float)HIDDEN);
            float var = ssq[row1] * (1.0f / (float)HIDDEN) - mu * mu;
            float v   = g * (c1[r] - mu) * rsqrtf(var + LN_EPS) + bb;
            int grow  = blockRow + row1;
            if (grow < N_NODES)
                xout[(size_t)grow * HIDDEN + col] = fmaxf(v, 0.0f);
        }
    }
}

// ---------------- graph mean pool (accumulate) ----------------
__global__ __launch_bounds__(256) void pool_kernel(const float* __restrict__ x,
                                                   const int* __restrict__ batch,
                                                   float* __restrict__ pooled,
                                                   float* __restrict__ cnt) {
    int gid  = blockIdx.x * blockDim.x + threadIdx.x;
    int node = gid >> 5;
    int lane = gid & 31;
    if (node >= N_NODES) return;
    int gph = batch[node];
    const float4 v = *(const float4*)(x + (size_t)node * HIDDEN + lane * 4);
    float* p = pooled + (size_t)gph * HIDDEN + lane * 4;
    atomicAdd(p + 0, v.x);
    atomicAdd(p + 1, v.y);
    atomicAdd(p + 2, v.z);
    atomicAdd(p + 3, v.w);
    if (lane == 0) atomicAdd(&cnt[gph], 1.0f);
}

// ---------------- classifier: (pooled/cnt) @ Wc + bc ----------------
__global__ void cls_kernel(const float* __restrict__ pooled,
                           const float* __restrict__ cnt,
                           const float* __restrict__ Wc,
                           const float* __restrict__ bc,
                           float* __restrict__ out) {
    int i = blockIdx.x * blockDim.x + threadIdx.x;
    if (i >= N_GRAPHS * OUTF) return;
    int gph = i / OUTF;
    int o   = i % OUTF;
    float inv = 1.0f / fmaxf(cnt[gph], 1.0f);
    float acc = bc[o];
    for (int f = 0; f < HIDDEN; ++f)
        acc += pooled[(size_t)gph * HIDDEN + f] * inv * Wc[(size_t)f * OUTF + o];
    out[i] = acc;
}

extern "C" void kernel_launch(void* const* d_in, const int* in_sizes, int n_in,
                              void* d_out, int out_size, void* d_ws, size_t ws_size,
                              hipStream_t stream) {
    (void)in_sizes; (void)n_in; (void)out_size; (void)ws_size;

    const float* x      = (const float*)d_in[0];
    const int*   eidx   = (const int*)d_in[1];   // [2, N_EDGES] flat
    const int*   batch  = (const int*)d_in[2];
    const float* W_rel  = (const float*)d_in[3]; // [2,128,128]
    const float* b_rel  = (const float*)d_in[4]; // [2,128]
    const float* W_root = (const float*)d_in[5]; // [2,128,128]
    const float* gamma  = (const float*)d_in[6]; // [2,128]
    const float* beta   = (const float*)d_in[7]; // [2,128]
    const float* Wc     = (const float*)d_in[8]; // [128,10]
    const float* bc     = (const float*)d_in[9]; // [10]
    float* out          = (float*)d_out;

    const int* src = eidx;
    const int* dst = eidx + N_EDGES;

    // workspace layout
    char* ws = (char*)d_ws;
    const size_t featBytes = (size_t)N_NODES * HIDDEN * sizeof(float); // 25.6 MB
    float* deg    = (float*)(ws);                                  // 50000 f32
    float* agg    = (float*)(ws + 204800);                         // [N,128]
    float* x1     = (float*)(ws + 204800 + featBytes);             // [N,128]
    float* x2     = (float*)(ws + 204800 + 2 * featBytes);         // [N,128]
    float* pooled = (float*)(ws + 204800 + 3 * featBytes);         // [128,128]
    float* cnt    = (float*)(ws + 204800 + 3 * featBytes + 65536); // [128]

    // zero the accumulation buffers (graph-capturable async memsets)
    hipMemsetAsync(deg, 0, (size_t)N_NODES * sizeof(float), stream);
    hipMemsetAsync(agg, 0, featBytes, stream);
    hipMemsetAsync(pooled, 0, (size_t)N_GRAPHS * HIDDEN * sizeof(float), stream);
    hipMemsetAsync(cnt, 0, (size_t)N_GRAPHS * sizeof(float), stream);

    // degree -> inverse degree (in place)
    deg_kernel<<<(N_EDGES + 255) / 256, 256, 0, stream>>>(dst, deg);
    invdeg_kernel<<<(N_NODES + 255) / 256, 256, 0, stream>>>(deg);

    const int scatterBlocks = (N_EDGES * 32) / 256;            // 100000
    const int gemmBlocks    = (N_NODES + MTILE - 1) / MTILE;   // 1563

    // ---- layer 0 ----
    scatter_kernel<<<scatterBlocks, 256, 0, stream>>>(x, src, dst, agg);
    gconv_kernel<<<gemmBlocks, 256, 0, stream>>>(
        x, agg, deg,
        W_rel, b_rel, W_root, gamma, beta, x1);

    // ---- layer 1 ----
    hipMemsetAsync(agg, 0, featBytes, stream);
    scatter_kernel<<<scatterBlocks, 256, 0, stream>>>(x1, src, dst, agg);
    gconv_kernel<<<gemmBlocks, 256, 0, stream>>>(
        x1, agg, deg,
        W_rel + HIDDEN * HIDDEN, b_rel + HIDDEN,
        W_root + HIDDEN * HIDDEN, gamma + HIDDEN, beta + HIDDEN, x2);

    // ---- pool + classify ----
    pool_kernel<<<(N_NODES * 32) / 256 + 1, 256, 0, stream>>>(x2, batch, pooled, cnt);
    cls_kernel<<<(N_GRAPHS * OUTF + 255) / 256, 256, 0, stream>>>(pooled, cnt, Wc, bc, out);
}